// NGP_42262478193063
// MI455X (gfx1250) — compile-verified
//
#include <hip/hip_runtime.h>
#include <cmath>
#include <cstdint>

#define NLVL 16
#define TSIZE (1u << 19)
#define LDS_FLOATS 19456   // 77,824 B: fits dense levels 0..5 (19,104 floats)
#define BLOCK 256

typedef __attribute__((ext_vector_type(2))) float v2f;
typedef __attribute__((ext_vector_type(4))) float v4f;
typedef __attribute__((ext_vector_type(4))) int   v4i;

#define AS1 __attribute__((address_space(1)))
#define AS3 __attribute__((address_space(3)))

#if __has_builtin(__builtin_amdgcn_global_load_async_to_lds_b128)
#define HAVE_ASYNC_LDS 1
#else
#define HAVE_ASYNC_LDS 0
#endif

struct NgpParams {
  const float* __restrict__ x;
  float* __restrict__ out;
  const float* __restrict__ tables[NLVL];
  float gs[NLVL];
  int   res[NLVL];
  int   hashed[NLVL];
  int   lds_off[NLVL];
  int   lds_sz[NLVL];
  int   lds_count;
  int   n;
};

__device__ __forceinline__ void async_b128(const float* g, float* l) {
#if HAVE_ASYNC_LDS
  // Param 1: AS1 v4i* (global src), param 2: AS3 v4i* (LDS dst), imm offset, cpol.
  __builtin_amdgcn_global_load_async_to_lds_b128((AS1 v4i*)(void*)g,
                                                 (AS3 v4i*)(void*)l, 0, 0);
#else
  *(v4f*)l = *(const v4f*)g;   // sync fallback: global_load_b128 + ds_store_b128
#endif
}

__device__ __forceinline__ void wait_async0() {
#if __has_builtin(__builtin_amdgcn_s_wait_asynccnt)
  __builtin_amdgcn_s_wait_asynccnt(0);
#else
  asm volatile("s_wait_asynccnt 0" ::: "memory");
#endif
}

// One level's bilinear interp for one point; writes o2[0..1].
__device__ __forceinline__ void level_eval(int res, int hashed, float gs,
                                           float px, float py,
                                           const float* __restrict__ tab,
                                           float* o2) {
  float fx = floorf(px / gs), fy = floorf(py / gs);
  int ix = (int)fx, iy = (int)fy;
  float wx = (px - fx * gs) / gs;
  float wy = (py - fy * gs) / gs;
  uint32_t i00, i01, i10, i11;
  if (hashed) {
    // h = cx * 1 ^ cy * P1 (int32 wrap), idx = h & (T-1)
    uint32_t a0 = (uint32_t)ix, a1 = (uint32_t)(ix + 1);
    uint32_t b0 = (uint32_t)iy * 2654435761u;
    uint32_t b1 = (uint32_t)(iy + 1) * 2654435761u;
    i00 = (a0 ^ b0) & (TSIZE - 1u);
    i01 = (a0 ^ b1) & (TSIZE - 1u);
    i10 = (a1 ^ b0) & (TSIZE - 1u);
    i11 = (a1 ^ b1) & (TSIZE - 1u);
  } else {
    uint32_t base = (uint32_t)(ix * res + iy);
    i00 = base;                  i01 = base + 1u;
    i10 = base + (uint32_t)res;  i11 = base + (uint32_t)res + 1u;
  }
  v2f e00 = ((const v2f*)tab)[i00];
  v2f e01 = ((const v2f*)tab)[i01];
  v2f e10 = ((const v2f*)tab)[i10];
  v2f e11 = ((const v2f*)tab)[i11];
  float uy = 1.0f - wy, ux = 1.0f - wx;
  float c00 = e00.x * uy + e01.x * wy;
  float c01 = e00.y * uy + e01.y * wy;
  float c10 = e10.x * uy + e11.x * wy;
  float c11 = e10.y * uy + e11.y * wy;
  o2[0] = c00 * ux + c10 * wx;
  o2[1] = c01 * ux + c11 * wx;
}

__global__ __launch_bounds__(BLOCK) void ngp_hashgrid_fwd(NgpParams p) {
  __shared__ float lut[LDS_FLOATS];
  const int tid = threadIdx.x;

  // ---- Kick off async staging of low-res dense tables into LDS ----
  for (int l = 0; l < p.lds_count; ++l) {
    const float* src = p.tables[l];
    float* dst = &lut[p.lds_off[l]];
    const int nf  = p.lds_sz[l];
    const int nf4 = nf & ~3;
    for (int i = tid * 4; i < nf4; i += BLOCK * 4)
      async_b128(src + i, dst + i);
    for (int i = nf4 + tid; i < nf; i += BLOCK)   // tail (<=3 floats)
      dst[i] = src[i];
  }

  const int  gid    = blockIdx.x * BLOCK + tid;
  const bool active = gid < p.n;
  float px = 0.0f, py = 0.0f;
  if (active) {
    const v2f pt = ((const v2f*)p.x)[gid];
    px = pt.x; py = pt.y;
  }

  float o[2 * NLVL];

  // ---- Phase 1: global-gather levels (overlaps with async LDS fill) ----
  if (active) {
#pragma unroll
    for (int l = 0; l < NLVL; ++l) {
      if (l < p.lds_count) continue;
      level_eval(p.res[l], p.hashed[l], p.gs[l], px, py, p.tables[l], &o[2 * l]);
    }
  }

  wait_async0();
  __syncthreads();

  // ---- Phase 2: LDS-resident levels + streaming NT store ----
  if (active) {
#pragma unroll
    for (int l = 0; l < NLVL; ++l) {
      if (l >= p.lds_count) continue;
      level_eval(p.res[l], /*hashed=*/0, p.gs[l], px, py, &lut[p.lds_off[l]], &o[2 * l]);
    }
    v4f* op = (v4f*)(p.out + (size_t)gid * (2 * NLVL));
#pragma unroll
    for (int j = 0; j < 8; ++j) {
      v4f v = { o[4 * j], o[4 * j + 1], o[4 * j + 2], o[4 * j + 3] };
      __builtin_nontemporal_store(v, op + j);
    }
  }
}

extern "C" void kernel_launch(void* const* d_in, const int* in_sizes, int n_in,
                              void* d_out, int out_size, void* d_ws, size_t ws_size,
                              hipStream_t stream) {
  (void)n_in; (void)out_size; (void)d_ws; (void)ws_size;

  NgpParams p{};
  p.x   = (const float*)d_in[0];
  p.out = (float*)d_out;
  p.n   = in_sizes[0] / 2;

  // Replicate numpy's float32 level config exactly (same libm on host).
  const float b = expf((logf(1024.0f) - logf(16.0f)) / 15.0f);
  int off = 0, count = 0;
  bool stop = false;
  for (int i = 0; i < NLVL; ++i) {
    p.tables[i] = (const float*)d_in[1 + i];
    const int res = (int)floorf(16.0f * powf(b, (float)i));
    p.res[i] = res;
    // hashed iff table size == T*F (dense sizes can never equal 2^20).
    const int sz = in_sizes[1 + i];
    p.hashed[i] = (sz == (int)(TSIZE * 2u)) ? 1 : 0;
    p.gs[i] = (float)(1024.0 / (double)res);
    if (!stop && !p.hashed[i] && off + sz <= LDS_FLOATS) {
      p.lds_off[i] = off;
      p.lds_sz[i]  = sz;
      off += sz;
      ++count;
    } else {
      stop = true;
      p.lds_off[i] = 0;
      p.lds_sz[i]  = 0;
    }
  }
  p.lds_count = count;

  const int blocks = (p.n + BLOCK - 1) / BLOCK;
  hipLaunchKernelGGL(ngp_hashgrid_fwd, dim3(blocks), dim3(BLOCK), 0, stream, p);
}